// QuantumCircuitSimulator_74741020885597
// MI455X (gfx1250) — compile-verified
//
#include <hip/hip_runtime.h>
#include <hip/hip_bf16.h>

typedef __attribute__((ext_vector_type(2))) float v2f;
typedef __attribute__((ext_vector_type(8))) float v8f;

// ---------------------------------------------------------------------------
// Kernel 1: compose each group of ladder gates into a fused 16x16 fp32 matrix.
// Group g (g=0..6) covers qubits [3g, 3g+3]; gates 3g..3g+2 (g<6) or 18..19
// (g==6, identity on local bit 3).  M_g = U_last * ... * U_first.
// ---------------------------------------------------------------------------
__global__ __launch_bounds__(256)
void build_fused(const float* __restrict__ gates, float* __restrict__ fused) {
    __shared__ float M[256];
    const int t  = threadIdx.x;
    const int g  = blockIdx.x;      // 0..6
    const int mp = t >> 4;          // output row m'
    const int c  = t & 15;          // column
    const int first  = 3 * g;
    const int ngates = (g < 6) ? 3 : 2;

    M[t] = (mp == c) ? 1.0f : 0.0f;
    __syncthreads();

    for (int jj = 0; jj < ngates; ++jj) {
        const int j = first + jj;   // global gate index
        const int p = jj;           // local qubit pair (p, p+1)
        // gate's 4-dim index for row m': bit0 = local bit p, bit1 = local bit p+1
        const int kp = (mp >> p) & 3;
        const int mb = mp & ~(3 << p);
        float acc = 0.0f;
        #pragma unroll
        for (int k = 0; k < 4; ++k)
            acc += gates[j * 16 + kp * 4 + k] * M[((mb | (k << p)) << 4) | c];
        __syncthreads();
        M[t] = acc;
        __syncthreads();
    }
    fused[g * 256 + t] = M[t];
}

// ---------------------------------------------------------------------------
// Kernel 2: apply a fused 16x16 matrix to the statevector on qubits
// [base, base+3].  One wave handles a 16(row-groups) x 16(local basis) tile:
//   D(16x16) = sum_{c=0..3} A_c(16x4) x B_c(4x16)   via v_wmma_f32_16x16x4_f32
// A rows   = 16 amplitude groups (row-group id rp = tile*16 + lane%16)
// B        = fused matrix, B_c[k][n] = M[n][4c+k]  (so D = V * M^T, out = M v)
// Address of (rp, j): insert j's 4 bits at bit position `base`.
// ---------------------------------------------------------------------------
__global__ __launch_bounds__(256)
void apply_fused16(const float* src, float* dst,
                   const float* __restrict__ M, int base) {
    const int lane  = threadIdx.x & 31;
    const int wave  = threadIdx.x >> 5;
    const int half  = lane >> 4;        // 0: K/M-half {0,1}/rows 0-7 ; 1: {2,3}/rows 8-15
    const int idx16 = lane & 15;
    const unsigned tile    = blockIdx.x * 8u + (unsigned)wave;   // 0..65535
    const unsigned lowmask = (1u << base) - 1u;

    // B operand per K-chunk: lane n holds M[n][4c+2*half], M[n][4c+2*half+1]
    v2f b[4];
    #pragma unroll
    for (int c = 0; c < 4; ++c) {
        const float* mptr = M + idx16 * 16 + 4 * c + 2 * half;
        b[c].x = mptr[0];
        b[c].y = mptr[1];
    }

    // A operand: lane holds row rp = tile*16 + idx16, K values {4c+2h, 4c+2h+1}
    const unsigned rp       = tile * 16u + (unsigned)idx16;
    const unsigned baseaddr = ((rp & ~lowmask) << 4) | (rp & lowmask);
    v2f a[4];
    #pragma unroll
    for (int c = 0; c < 4; ++c) {
        const unsigned j0 = (unsigned)(4 * c + 2 * half);
        a[c].x = src[baseaddr + (j0 << base)];
        a[c].y = src[baseaddr + ((j0 + 1u) << base)];
    }

    // Accumulate D = sum_c A_c x B_c   (fp32 WMMA, full precision)
    v8f acc = {};
    #pragma unroll
    for (int c = 0; c < 4; ++c) {
        acc = __builtin_amdgcn_wmma_f32_16x16x4_f32(
            /*neg_a=*/false, a[c], /*neg_b=*/false, b[c],
            /*c_mod=*/(short)0, acc, /*reuse_a=*/false, /*reuse_b=*/false);
    }

    // D layout: VGPR v -> row (v + 8*half), column idx16
    #pragma unroll
    for (int v = 0; v < 8; ++v) {
        const unsigned r  = tile * 16u + (unsigned)(v + 8 * half);
        const unsigned ad = ((r & ~lowmask) << 4) |
                            ((unsigned)idx16 << base) | (r & lowmask);
        dst[ad] = acc[v];
    }
}

// ---------------------------------------------------------------------------
// kernel_launch: build 7 fused matrices, then 7 state-wide passes.
// Pass 0: d_in -> d_out ; passes 1..6: in-place on d_out (disjoint 16-groups
// within a pass; kernel boundaries provide the global sync between passes).
// ---------------------------------------------------------------------------
extern "C" void kernel_launch(void* const* d_in, const int* in_sizes, int n_in,
                              void* d_out, int out_size, void* d_ws, size_t ws_size,
                              hipStream_t stream) {
    const float* state = (const float*)d_in[0];   // 2^24 fp32
    const float* gates = (const float*)d_in[1];   // 20 x 4 x 4 fp32
    float* out   = (float*)d_out;
    float* fused = (float*)d_ws;                  // 7 * 256 fp32 = 7 KB

    build_fused<<<7, 256, 0, stream>>>(gates, fused);

    // 2^24 amplitudes / 256 per wave-tile = 65536 waves = 8192 blocks of 8 waves
    apply_fused16<<<8192, 256, 0, stream>>>(state, out, fused, 0);
    for (int g = 1; g < 7; ++g) {
        apply_fused16<<<8192, 256, 0, stream>>>(out, out, fused + g * 256, 3 * g);
    }
}